// TDHPB_Layer_19327352832046
// MI455X (gfx1250) — compile-verified
//
#include <hip/hip_runtime.h>
#include <hip/hip_bf16.h>
#include <math.h>

// Problem constants (from reference)
#define BB 48
#define TT 4096
#define DD 256
#define SS 10          // N_STAGES
#define BLK 455        // BLOCK
#define SCALE 0.0625f  // 1/sqrt(256)

typedef float v2f __attribute__((ext_vector_type(2)));
typedef float v8f __attribute__((ext_vector_type(8)));

__device__ __forceinline__ float4 max4(float4 a, float4 b) {
    float4 r;
    r.x = fmaxf(a.x, b.x); r.y = fmaxf(a.y, b.y);
    r.z = fmaxf(a.z, b.z); r.w = fmaxf(a.w, b.w);
    return r;
}

// ---------------------------------------------------------------------------
// Kernel 1: block summaries q[b,s,d] = max over chunk rows (zero-padded tail)
// grid = B*S blocks, 256 threads. Thread t owns 4 consecutive columns
// (float4 = global_load_b128); 4 row-groups combined via LDS.
// ---------------------------------------------------------------------------
__global__ void k_block_max(const float* __restrict__ x, float* __restrict__ q) {
    __shared__ float4 sm4[256];

    const int bs = blockIdx.x;            // b*SS + s
    const int b  = bs / SS;
    const int s  = bs % SS;
    const int t  = threadIdx.x;
    const int g  = t >> 6;                // row group 0..3
    const int c4 = (t & 63) * 4;          // first of 4 owned columns
    const int base = s * BLK;
    const size_t xbase = ((size_t)b * TT) * DD;

    float4 m = { -INFINITY, -INFINITY, -INFINITY, -INFINITY };
    for (int r = g; r < BLK; r += 4) {
        const int row = base + r;
        float4 v = { 0.0f, 0.0f, 0.0f, 0.0f };       // zero-pad tail rows
        if (row < TT)
            v = *reinterpret_cast<const float4*>(x + xbase + (size_t)row * DD + c4);
        m = max4(m, v);
    }
    sm4[t] = m;
    __syncthreads();
    if (g == 0) {
        float4 a = sm4[t], bb = sm4[t + 64], c = sm4[t + 128], d = sm4[t + 192];
        float4 r = max4(max4(a, bb), max4(c, d));
        *reinterpret_cast<float4*>(q + (size_t)bs * DD + c4) = r;
    }
}

// ---------------------------------------------------------------------------
// Kernel 2: q_block = softmax(q q^T / 16) q, per batch. One wave32 per batch.
// V_WMMA_F32_16X16X4_F32, 10 rows padded to 16.
// ---------------------------------------------------------------------------
__global__ void k_qblock_attn(const float* __restrict__ q,
                              float* __restrict__ q_block) {
    __shared__ float qs[16][260];
    __shared__ float ps[16][17];

    const int b    = blockIdx.x;
    const int lane = threadIdx.x;   // 0..31
    const int m    = lane & 15;
    const int hi   = lane >> 4;

    for (int i = lane; i < 16 * DD; i += 32) {
        const int r  = i >> 8;
        const int dd = i & 255;
        qs[r][dd] = (r < SS) ? q[((size_t)b * SS + r) * DD + dd] : 0.0f;
    }
    __syncthreads();

    // ---- S = Q * Q^T : identical A/B register images (see layout analysis) ----
    v8f c = {};
    #pragma unroll
    for (int k0 = 0; k0 < DD; k0 += 4) {
        v2f a;
        a.x = qs[m][k0 + 2 * hi];
        a.y = qs[m][k0 + 1 + 2 * hi];
        c = __builtin_amdgcn_wmma_f32_16x16x4_f32(false, a, false, a,
                                                  (short)0, c, false, false);
    }

    // ---- row softmax. C layout: vgpr v, lane -> S[v+8*hi][lane&15] ----
    v8f p;
    #pragma unroll
    for (int v = 0; v < 8; ++v) {
        float sv = c[v] * SCALE;
        if (m >= SS) sv = -INFINITY;            // mask padded key columns
        float mx = sv;
        mx = fmaxf(mx, __shfl_xor(mx, 1));
        mx = fmaxf(mx, __shfl_xor(mx, 2));
        mx = fmaxf(mx, __shfl_xor(mx, 4));
        mx = fmaxf(mx, __shfl_xor(mx, 8));
        float e = __expf(sv - mx);
        float sum = e;
        sum += __shfl_xor(sum, 1);
        sum += __shfl_xor(sum, 2);
        sum += __shfl_xor(sum, 4);
        sum += __shfl_xor(sum, 8);
        p[v] = e / sum;
    }

    #pragma unroll
    for (int v = 0; v < 8; ++v) ps[v + 8 * hi][m] = p[v];
    __syncthreads();

    // ---- O = P * V ----
    #pragma unroll 4
    for (int n0 = 0; n0 < DD; n0 += 16) {
        v8f acc = {};
        #pragma unroll
        for (int k0 = 0; k0 < 16; k0 += 4) {
            v2f a, bmat;
            a.x    = ps[m][k0 + 2 * hi];
            a.y    = ps[m][k0 + 1 + 2 * hi];
            bmat.x = qs[k0 + 2 * hi][n0 + m];
            bmat.y = qs[k0 + 1 + 2 * hi][n0 + m];
            acc = __builtin_amdgcn_wmma_f32_16x16x4_f32(false, a, false, bmat,
                                                        (short)0, acc, false, false);
        }
        #pragma unroll
        for (int v = 0; v < 8; ++v) {
            const int M = v + 8 * hi;
            if (M < SS)
                q_block[((size_t)b * SS + M) * DD + n0 + m] = acc[v];
        }
    }
}

// ---------------------------------------------------------------------------
// Kernel 3: q_step = softmax(q_block . chunk^T / 16) . chunk, per (b,s).
// grid = B*S blocks, 256 threads (8 waves). Memory bound (M=1 -> no WMMA).
// ---------------------------------------------------------------------------
__global__ void k_qstep(const float* __restrict__ x,
                        const float* __restrict__ q_block,
                        float* __restrict__ q_step) {
    __shared__ float qb[DD];
    __shared__ float sc[BLK + 1];
    __shared__ float red[256];
    __shared__ float4 sm4[256];

    const int bs = blockIdx.x;
    const int b  = bs / SS;
    const int s  = bs % SS;
    const int t  = threadIdx.x;
    const int w    = t >> 5;
    const int lane = t & 31;
    const int lim  = min(BLK, TT - s * BLK);

    qb[t] = q_block[(size_t)bs * DD + t];
    __syncthreads();

    // Phase B: scores. Wave w handles keys w, w+8,...; lane covers 8 dims.
    const size_t chunk_base = ((size_t)b * TT + (size_t)s * BLK) * DD;
    for (int j = w; j < BLK; j += 8) {
        float partial = 0.0f;
        if (j < lim) {
            const float* rowp = x + chunk_base + (size_t)j * DD + lane * 8;
            const float4 r0 = reinterpret_cast<const float4*>(rowp)[0];
            const float4 r1 = reinterpret_cast<const float4*>(rowp)[1];
            const int d0 = lane * 8;
            partial = r0.x * qb[d0]     + r0.y * qb[d0 + 1] +
                      r0.z * qb[d0 + 2] + r0.w * qb[d0 + 3] +
                      r1.x * qb[d0 + 4] + r1.y * qb[d0 + 5] +
                      r1.z * qb[d0 + 6] + r1.w * qb[d0 + 7];
        }
        #pragma unroll
        for (int off = 16; off > 0; off >>= 1) partial += __shfl_xor(partial, off);
        if (lane == 0) sc[j] = partial * SCALE;   // pad rows -> score 0
    }
    __syncthreads();

    // Phase C: block softmax over all BLK scores (pads included)
    float lm = -INFINITY;
    for (int j = t; j < BLK; j += 256) lm = fmaxf(lm, sc[j]);
    red[t] = lm;
    __syncthreads();
    for (int o = 128; o > 0; o >>= 1) {
        if (t < o) red[t] = fmaxf(red[t], red[t + o]);
        __syncthreads();
    }
    const float mx = red[0];
    __syncthreads();

    float ls = 0.0f;
    for (int j = t; j < BLK; j += 256) {
        const float e = __expf(sc[j] - mx);
        sc[j] = e;
        ls += e;
    }
    red[t] = ls;
    __syncthreads();
    for (int o = 128; o > 0; o >>= 1) {
        if (t < o) red[t] += red[t + o];
        __syncthreads();
    }
    const float inv = 1.0f / red[0];
    __syncthreads();

    // Phase D: o[d] = sum_j p[j]*x[row_j][d]; thread owns 4 cols (b128 loads),
    // 4 row-groups combined through LDS.
    const int g  = t >> 6;
    const int c4 = (t & 63) * 4;
    float4 o = { 0.0f, 0.0f, 0.0f, 0.0f };
    for (int j = g; j < lim; j += 4) {
        const float4 v = *reinterpret_cast<const float4*>(
            x + chunk_base + (size_t)j * DD + c4);
        const float pj = sc[j];
        o.x += pj * v.x; o.y += pj * v.y; o.z += pj * v.z; o.w += pj * v.w;
    }
    sm4[t] = o;
    __syncthreads();
    if (g == 0) {
        float4 a = sm4[t], bb = sm4[t + 64], cc = sm4[t + 128], d = sm4[t + 192];
        float4 r;
        r.x = ((a.x + bb.x) + (cc.x + d.x)) * inv;
        r.y = ((a.y + bb.y) + (cc.y + d.y)) * inv;
        r.z = ((a.z + bb.z) + (cc.z + d.z)) * inv;
        r.w = ((a.w + bb.w) + (cc.w + d.w)) * inv;
        *reinterpret_cast<float4*>(q_step + (size_t)bs * DD + c4) = r;
    }
}

// ---------------------------------------------------------------------------
// Kernel 4: out = concat(q_block, q_step) @ W^T + bias.
// One wave per 16x16 tile; A/B tiles staged in LDS with coalesced float4
// loads, then K=512 -> 128 WMMA fed from LDS (ds_load_2addr pipeline).
// ---------------------------------------------------------------------------
__global__ void k_fuse(const float* __restrict__ q_block,
                       const float* __restrict__ q_step,
                       const float* __restrict__ Wf,   // [256, 512]
                       const float* __restrict__ bf,   // [256]
                       float* __restrict__ out) {      // [480, 256]
    __shared__ float As[16][520];   // fused rows  [16 x 512]
    __shared__ float Bs[16][520];   // W rows (out cols) [16 x 512]

    const int tile = blockIdx.x;
    const int row0 = (tile >> 4) * 16;
    const int col0 = (tile & 15) * 16;
    const int lane = threadIdx.x;
    const int m    = lane & 15;
    const int hi   = lane >> 4;

    // Cooperative coalesced tile loads: 16*512/4 = 2048 float4s, 32 lanes.
    for (int i = lane; i < 16 * 512 / 4; i += 32) {
        const int r    = i >> 7;          // /128 float4s per row
        const int col  = (i & 127) * 4;
        // A: fused[row0+r][col] = col<256 ? q_block : q_step
        float4 av;
        if (col < 256)
            av = *reinterpret_cast<const float4*>(
                q_block + (size_t)(row0 + r) * DD + col);
        else
            av = *reinterpret_cast<const float4*>(
                q_step + (size_t)(row0 + r) * DD + (col - 256));
        *reinterpret_cast<float4*>(&As[r][col]) = av;
        // B: W[col0+r][col]
        const float4 bv = *reinterpret_cast<const float4*>(
            Wf + (size_t)(col0 + r) * 512 + col);
        *reinterpret_cast<float4*>(&Bs[r][col]) = bv;
    }
    __syncthreads();

    v8f acc = {};
    #pragma unroll 8
    for (int k0 = 0; k0 < 512; k0 += 4) {
        const int ka = k0 + 2 * hi;
        v2f a, bmat;
        a.x    = As[m][ka];
        a.y    = As[m][ka + 1];
        bmat.x = Bs[m][ka];
        bmat.y = Bs[m][ka + 1];
        acc = __builtin_amdgcn_wmma_f32_16x16x4_f32(false, a, false, bmat,
                                                    (short)0, acc, false, false);
    }

    const float bias = bf[col0 + m];
    #pragma unroll
    for (int v = 0; v < 8; ++v) {
        const int r = row0 + v + 8 * hi;
        out[(size_t)r * DD + col0 + m] = acc[v] + bias;
    }
}

// ---------------------------------------------------------------------------
extern "C" void kernel_launch(void* const* d_in, const int* in_sizes, int n_in,
                              void* d_out, int out_size, void* d_ws, size_t ws_size,
                              hipStream_t stream) {
    (void)in_sizes; (void)n_in; (void)out_size; (void)ws_size;
    const float* x  = (const float*)d_in[0];   // [48,4096,256]
    const float* Wf = (const float*)d_in[1];   // [256,512]
    const float* bf = (const float*)d_in[2];   // [256]
    float* out = (float*)d_out;                // [48,10,256]

    float* ws       = (float*)d_ws;
    float* q        = ws;                              // [480,256]
    float* q_block  = ws + (size_t)BB * SS * DD;       // [480,256]
    float* q_step   = ws + (size_t)2 * BB * SS * DD;   // [480,256]

    k_block_max  <<<BB * SS, 256, 0, stream>>>(x, q);
    k_qblock_attn<<<BB, 32, 0, stream>>>(q, q_block);
    k_qstep      <<<BB * SS, 256, 0, stream>>>(x, q_block, q_step);
    k_fuse       <<<(BB * SS / 16) * (DD / 16), 32, 0, stream>>>(q_block, q_step, Wf, bf, out);
}